// SelfAttn_37099927503247
// MI455X (gfx1250) — compile-verified
//
#include <hip/hip_runtime.h>
#include <math.h>

// Problem constants (match reference)
#define NB   16
#define MM   1024
#define EMBD 512
#define OUTD 512
#define SROW 1025   // LDS row stride (1024 + 1 pad -> conflict-free strided reads)

typedef __attribute__((ext_vector_type(2))) float v2f;
typedef __attribute__((ext_vector_type(4))) float v4f;
typedef __attribute__((ext_vector_type(8))) float v8f;

static __device__ __forceinline__ v8f wmma4(v2f a, v2f b, v8f c) {
    // D = A(16x4 f32) x B(4x16 f32) + C(16x16 f32), full fp32 precision
    return __builtin_amdgcn_wmma_f32_16x16x4_f32(
        /*neg_a=*/false, a, /*neg_b=*/false, b,
        /*c_mod=*/(short)0, c, /*reuse_a=*/false, /*reuse_b=*/false);
}

// ---------------------------------------------------------------------------
// Kernel 1: Q = joint*Wq, K = joint*Wk, V = joint*Wv   (rows = N*M = 16384)
// One wave computes one 16x16 tile of Q, K, V (A-fragment shared 3 ways).
// joint / W stay temporal (joint is reused by all 32 col-blocks -> L2 hit).
// grid = (OUTD/16, N*M/128), block = 256 (8 waves, wave w -> rows +16*w)
// ---------------------------------------------------------------------------
__global__ __launch_bounds__(256)
void qkv_kernel(const float* __restrict__ joint,
                const float* __restrict__ Wq,
                const float* __restrict__ Wk,
                const float* __restrict__ Wv,
                float* __restrict__ Q,
                float* __restrict__ K,
                float* __restrict__ V) {
    const int lane = threadIdx.x & 31;
    const int w    = threadIdx.x >> 5;
    const int l16  = lane & 15;
    const int h    = lane >> 4;          // half-wave select
    const int r0   = blockIdx.y * 128 + w * 16;
    const int c0   = blockIdx.x * 16;

    v8f cq = {}; v8f ck = {}; v8f cv = {};

    // A fragment: row = l16, k = kk + 2*h + {0,1}  (contiguous -> b64 load)
    const float* arow = joint + (size_t)(r0 + l16) * EMBD + 2 * h;

    for (int kk = 0; kk < EMBD; kk += 4) {
        v2f a = *(const v2f*)(arow + kk);
        const int br  = kk + 2 * h;                 // B row for v=0
        const size_t i0 = (size_t)br * OUTD + c0 + l16;
        const size_t i1 = i0 + OUTD;                // B row for v=1
        v2f bq; bq.x = Wq[i0]; bq.y = Wq[i1];
        v2f bk; bk.x = Wk[i0]; bk.y = Wk[i1];
        v2f bv; bv.x = Wv[i0]; bv.y = Wv[i1];
        cq = wmma4(a, bq, cq);
        ck = wmma4(a, bk, ck);
        cv = wmma4(a, bv, cv);
    }

    // C/D layout: row = v + 8*h, col = l16. Regular (temporal) stores: Q/K/V
    // are re-read by kernel 2 and should stay dirty/resident in L2 (96 MB of
    // 192 MB).
#pragma unroll
    for (int v = 0; v < 8; ++v) {
        const size_t idx = (size_t)(r0 + v + 8 * h) * OUTD + c0 + l16;
        Q[idx] = cq[v];
        K[idx] = ck[v];
        V[idx] = cv[v];
    }
}

// ---------------------------------------------------------------------------
// Kernel 2: fused  S = Q Kt + sum(delta)  -> softmax(full row) -> post-mask
//           -> out = P * V.  One block per (n, 16-row i-strip).
// 8 waves: phase 1 each wave owns a 128-wide j-range; phase 3 each wave owns
// a 64-wide o-slice. 16x1024 f32 score strip lives in (dynamic) LDS.
// delta (256 MB, read-once) is loaded non-temporal so it cannot evict the
// reused K/V/Q working set from L2. grid = (MM/16, NB), block = 256.
// ---------------------------------------------------------------------------
__global__ __launch_bounds__(256)
void attn_kernel(const float* __restrict__ Q,
                 const float* __restrict__ K,
                 const float* __restrict__ V,
                 const float* __restrict__ delta,
                 const int*   __restrict__ traj_len,
                 float* __restrict__ out) {
    extern __shared__ float smem[];
    float* Sbuf   = smem;                 // 16 * SROW
    float* red    = Sbuf + 16 * SROW;     // 16 x 16 partials
    float* rowm   = red + 256;            // 16 row maxima
    float* rscale = rowm + 16;            // 16 (valid_i / l_i)

    const int n   = blockIdx.y;
    const int i0  = blockIdx.x * 16;
    const int tid = threadIdx.x;
    const int lane = tid & 31, w = tid >> 5, l16 = lane & 15, h = lane >> 4;
    const int len = traj_len[n];

    const float* Qn = Q + (size_t)n * MM * OUTD;
    const float* Kn = K + (size_t)n * MM * OUTD;
    const float* Vn = V + (size_t)n * MM * OUTD;

    // ---------------- Phase 1: S strip = Q Kt + sum_k(delta) ----------------
    const float* qrow = Qn + (size_t)(i0 + l16) * OUTD + 2 * h;
    for (int jc = 0; jc < 8; ++jc) {
        const int j0 = w * 128 + jc * 16;
        const float* krow = Kn + (size_t)(j0 + l16) * OUTD + 2 * h;
        v8f c = {};
        for (int kk = 0; kk < OUTD; kk += 4) {
            v2f a = *(const v2f*)(qrow + kk);   // Q rows (A)
            v2f b = *(const v2f*)(krow + kk);   // Kt cols (B)
            c = wmma4(a, b, c);
        }
#pragma unroll
        for (int v = 0; v < 8; ++v) {
            const int i = i0 + v + 8 * h;
            const int j = j0 + l16;
            // 256 MB single-use stream: non-temporal b128 load (TH=NT)
            const v4f d4 = __builtin_nontemporal_load(
                (const v4f*)(delta + ((((size_t)n * MM + i) * MM + j) << 2)));
            Sbuf[(v + 8 * h) * SROW + j] = c[v] + d4.x + d4.y + d4.z + d4.w;
        }
    }
    __syncthreads();

    // ---------------- Phase 2: softmax over full row, then mask -------------
    {   // row-max partials: thread = (row r, 64-wide segment seg)
        const int r = tid & 15, seg = tid >> 4;
        const float* srow = Sbuf + r * SROW + seg * 64;
        float m = -3.402823466e+38f;
        for (int s = 0; s < 64; ++s) m = fmaxf(m, srow[s]);
        red[r * 16 + seg] = m;
    }
    __syncthreads();
    if (tid < 16) {
        float m = red[tid * 16];
        for (int s = 1; s < 16; ++s) m = fmaxf(m, red[tid * 16 + s]);
        rowm[tid] = m;
    }
    __syncthreads();
    {   // exp in place + sum partials
        const int r = tid & 15, seg = tid >> 4;
        const float m = rowm[r];
        float* srow = Sbuf + r * SROW + seg * 64;
        float acc = 0.f;
        for (int s = 0; s < 64; ++s) {
            const float e = __expf(srow[s] - m);
            srow[s] = e;
            acc += e;
        }
        red[r * 16 + seg] = acc;
    }
    __syncthreads();
    if (tid < 16) {
        float l = 0.f;
        for (int s = 0; s < 16; ++s) l += red[tid * 16 + s];
        const float vi = ((i0 + tid) < len) ? 1.f : 0.f;   // post-softmax row mask
        rscale[tid] = vi / l;
    }
    __syncthreads();
    {   // fold (valid_i * valid_j / l) into the strip
        const int r = tid & 15, seg = tid >> 4;
        const float sc = rscale[r];
        float* srow = Sbuf + r * SROW + seg * 64;
        const int jb = seg * 64;
        for (int s = 0; s < 64; ++s) {
            const float vj = ((jb + s) < len) ? 1.f : 0.f;
            srow[s] *= sc * vj;
        }
    }
    __syncthreads();

    // ---------------- Phase 3: out strip = P * V, wave w owns 64 o-cols -----
    const int oc0 = w * 64;
    v8f a0 = {}, a1 = {}, a2 = {}, a3 = {};
    for (int jc = 0; jc < 64; ++jc) {
#pragma unroll
        for (int ks = 0; ks < 4; ++ks) {
            const int kb = jc * 16 + ks * 4;
            v2f a;                                  // P fragment from LDS
            a.x = Sbuf[l16 * SROW + kb + 2 * h];
            a.y = Sbuf[l16 * SROW + kb + 2 * h + 1];
            const float* v0 = Vn + (size_t)(kb + 2 * h) * OUTD + oc0 + l16;
            const float* v1 = v0 + OUTD;
            v2f b0; b0.x = v0[0];  b0.y = v1[0];
            v2f b1; b1.x = v0[16]; b1.y = v1[16];
            v2f b2; b2.x = v0[32]; b2.y = v1[32];
            v2f b3; b3.x = v0[48]; b3.y = v1[48];
            a0 = wmma4(a, b0, a0);
            a1 = wmma4(a, b1, a1);
            a2 = wmma4(a, b2, a2);
            a3 = wmma4(a, b3, a3);
        }
    }
    // Output is write-once, never re-read: non-temporal stores.
    float* outn = out + (size_t)n * MM * OUTD;
#pragma unroll
    for (int v = 0; v < 8; ++v) {
        float* orow = outn + (size_t)(i0 + v + 8 * h) * OUTD + oc0 + l16;
        __builtin_nontemporal_store(a0[v], orow);
        __builtin_nontemporal_store(a1[v], orow + 16);
        __builtin_nontemporal_store(a2[v], orow + 32);
        __builtin_nontemporal_store(a3[v], orow + 48);
    }
}

// ---------------------------------------------------------------------------
extern "C" void kernel_launch(void* const* d_in, const int* in_sizes, int n_in,
                              void* d_out, int out_size, void* d_ws, size_t ws_size,
                              hipStream_t stream) {
    (void)in_sizes; (void)n_in; (void)out_size; (void)ws_size;
    const float* joint = (const float*)d_in[0];
    const float* delta = (const float*)d_in[1];
    const int*   traj  = (const int*)  d_in[2];
    const float* Wq    = (const float*)d_in[3];
    const float* Wk    = (const float*)d_in[4];
    const float* Wv    = (const float*)d_in[5];
    float* out = (float*)d_out;

    const size_t mat = (size_t)NB * MM * OUTD;   // 8M floats = 32 MB each
    float* Q = (float*)d_ws;
    float* K = Q + mat;
    float* V = K + mat;

    dim3 g1(OUTD / 16, (NB * MM) / 128);         // (32, 128)
    qkv_kernel<<<g1, 256, 0, stream>>>(joint, Wq, Wk, Wv, Q, K, V);

    const size_t lds = (size_t)(16 * SROW + 256 + 16 + 16) * sizeof(float); // ~65.2 KB
    dim3 g2(MM / 16, NB);                        // (64, 16)
    attn_kernel<<<g2, 256, lds, stream>>>(Q, K, V, delta, traj, out);
}